// DecodePredictions_39986145525864
// MI455X (gfx1250) — compile-verified
//
#include <hip/hip_runtime.h>
#include <hip/hip_bf16.h>
#include <math.h>

#define NUM_CLASSES 20
#define NANCH       76725
#define KTOP        100
#define MAXDET      100
#define CHUNK       38400
#define GS          64
#define NGRP        (CHUNK / GS)      // 600
#define SENT        (-3.0e38f)

typedef __attribute__((ext_vector_type(2))) float v2f;
typedef __attribute__((ext_vector_type(8))) float v8f;

#if defined(__has_builtin)
#  if __has_builtin(__builtin_amdgcn_s_wait_asynccnt)
#    define WAIT_ASYNC() __builtin_amdgcn_s_wait_asynccnt(0)
#  endif
#  if __has_builtin(__builtin_amdgcn_wmma_f32_16x16x4_f32)
#    define HAS_WMMA4 1
#  endif
#endif
#ifndef WAIT_ASYNC
#  define WAIT_ASYNC() asm volatile("s_wait_asynccnt 0" ::: "memory")
#endif

__device__ __forceinline__ void wave_argmax2(float& v, int& i) {
#pragma unroll
  for (int off = 16; off > 0; off >>= 1) {
    float ov = __shfl_down(v, off, 32);
    int   oi = __shfl_down(i, off, 32);
    if (ov > v || (ov == v && oi < i)) { v = ov; i = oi; }
  }
}

__device__ __forceinline__ void wave_argmax3(float& v, int& g, int& p) {
#pragma unroll
  for (int off = 16; off > 0; off >>= 1) {
    float ov = __shfl_down(v, off, 32);
    int   og = __shfl_down(g, off, 32);
    int   op = __shfl_down(p, off, 32);
    if (ov > v || (ov == v && og < g)) { v = ov; g = og; p = op; }
  }
}

// One workgroup per (class, batch): async-stage score column into LDS,
// tournament top-100 (reference tie semantics), decode selected boxes from
// analytic anchors, IoU NMS (union via WMMA outer-sum), emit candidates.
__global__ __launch_bounds__(256) void retina_perclass_nms(
    const float* __restrict__ box_pred, const float* __restrict__ cls_pred,
    float* __restrict__ cand_box, float* __restrict__ cand_score)
{
  __shared__ float s_buf[CHUNK];     // 150 KB staging; reused for boxes/area/iou
  __shared__ float s_cv[256];        // 2*100 chunk candidates (+pad)
  __shared__ int   s_cg[256];
  __shared__ float s_selv[128];      // final 100 selected (logit key or -1)
  __shared__ int   s_selg[128];      // global anchor index
  __shared__ float s_gv[NGRP];       // tournament group maxima
  __shared__ int   s_gi[NGRP];
  __shared__ int   s_keep[128];

  const int c   = blockIdx.x;
  const int b   = blockIdx.y;
  const int tid = threadIdx.x;
  const int nt  = 256;
  const int lane = tid & 31, wid = tid >> 5;

  const float* col = cls_pred + (size_t)b * NANCH * NUM_CLASSES + c;

  if (tid >= 2 * KTOP) { s_cv[tid] = SENT; s_cg[tid] = 0x7fffffff; }

  for (int ch = 0; ch < 2; ++ch) {
    const int base = ch * CHUNK;
    const int len  = (NANCH - base < CHUNK) ? (NANCH - base) : CHUNK;

    // ---- async stage: strided class column -> LDS (ASYNCcnt path) ----
    for (int i = tid; i < len; i += nt) {
      unsigned lds_addr = (unsigned)(size_t)&s_buf[i];
      unsigned long long gaddr =
          (unsigned long long)(size_t)(col + (size_t)(base + i) * NUM_CLASSES);
      asm volatile("global_load_async_to_lds_b32 %0, %1, off"
                   :: "v"(lds_addr), "v"(gaddr) : "memory");
    }
    WAIT_ASYNC();
    __syncthreads();

    // threshold map: sigmoid(x)>0.5 <=> x>0 ; below-threshold -> -1 (ref), pad -> SENT
    for (int i = tid; i < CHUNK; i += nt) {
      float v = (i < len) ? s_buf[i] : SENT;
      s_buf[i] = (v > 0.0f) ? v : ((i < len) ? -1.0f : SENT);
    }
    __syncthreads();

    // ---- build group maxima: one wave per group (conflict-free LDS) ----
    for (int g = wid; g < NGRP; g += 8) {
      const int g0 = g * GS;
      const int i0 = g0 + lane, i1 = g0 + 32 + lane;
      float v0 = s_buf[i0], v1 = s_buf[i1];
      float nv; int ni;
      if (v0 >= v1) { nv = v0; ni = i0; } else { nv = v1; ni = i1; }  // i0<i1
      wave_argmax2(nv, ni);
      if (lane == 0) { s_gv[g] = nv; s_gi[g] = ni; }
    }
    __syncthreads();

    // ---- tournament top-100 of this chunk (wave 0 only, no block barriers) ----
    if (wid == 0) {
      for (int k = 0; k < KTOP; ++k) {
        float bv = SENT; int bi = 0x7fffffff;
        for (int g = lane; g < NGRP; g += 32) {
          float v = s_gv[g]; int i = s_gi[g];
          if (v > bv || (v == bv && i < bi)) { bv = v; bi = i; }
        }
        wave_argmax2(bv, bi);
        bv = __shfl(bv, 0, 32);
        bi = __shfl(bi, 0, 32);
        if (lane == 0) {
          s_cv[ch * KTOP + k] = bv;
          s_cg[ch * KTOP + k] = (bi < CHUNK) ? (base + bi) : 0x7fffffff;
          if (bi < CHUNK) s_buf[bi] = SENT;      // consume winner
        }
        // recompute the one group that lost its winner
        const int g0 = (bi < CHUNK) ? (bi / GS) * GS : 0;
        const int i0 = g0 + lane, i1 = g0 + 32 + lane;
        float v0 = s_buf[i0], v1 = s_buf[i1];
        float nv; int ni;
        if (v0 >= v1) { nv = v0; ni = i0; } else { nv = v1; ni = i1; }
        wave_argmax2(nv, ni);
        if (lane == 0 && bi < CHUNK) { s_gv[g0 / GS] = nv; s_gi[g0 / GS] = ni; }
      }
    }
    __syncthreads();   // protect s_buf before next chunk's async staging
  }

  // ---- merge 200 chunk candidates -> final top-100 (tie -> smaller anchor idx) ----
  if (wid == 0) {
    for (int k = 0; k < KTOP; ++k) {
      float bv = SENT; int bg = 0x7fffffff; int bp2 = -1;
      for (int p = lane; p < 256; p += 32) {
        float v = s_cv[p]; int g = s_cg[p];
        if (v > bv || (v == bv && g < bg)) { bv = v; bg = g; bp2 = p; }
      }
      wave_argmax3(bv, bg, bp2);
      if (lane == 0) {
        s_selv[k] = bv; s_selg[k] = bg;
        if (bp2 >= 0) s_cv[bp2] = SENT;
      }
    }
  }
  __syncthreads();

  // ---- reuse staging LDS: boxes / areas / IoU matrix ----
  float* s_boxes = s_buf;          // [112][4]
  float* s_area  = s_buf + 512;    // [112]
  float* s_iou   = s_buf + 640;    // [112*112] (pitch 112)

  if (tid < 112) {
    if (tid < KTOP) {
      int idx = s_selg[tid];
      if (idx < 0 || idx >= NANCH) idx = 0;
      int basei, dim; float stride, area;
      if      (idx < 57600) { basei = 0;     dim = 80; stride = 8.f;   area = 1024.f;   }
      else if (idx < 72000) { basei = 57600; dim = 40; stride = 16.f;  area = 4096.f;   }
      else if (idx < 75600) { basei = 72000; dim = 20; stride = 32.f;  area = 16384.f;  }
      else if (idx < 76500) { basei = 75600; dim = 10; stride = 64.f;  area = 65536.f;  }
      else                  { basei = 76500; dim = 5;  stride = 128.f; area = 262144.f; }
      int r = idx - basei;
      int cell = r / 9, a = r - 9 * cell;
      int x = cell % dim, y = cell / dim;
      float cx = (x + 0.5f) * stride, cy = (y + 0.5f) * stride;
      int ri = a / 3, si = a - 3 * ri;
      float ratio = (ri == 0) ? 0.5f : ((ri == 1) ? 1.0f : 2.0f);
      float ah = sqrtf(area / ratio);
      float aw = area / ah;
      float sc = (si == 0) ? 1.0f
               : ((si == 1) ? 1.2599210498948732f : 1.5874010519681994f);
      aw *= sc; ah *= sc;
      const float4 t = *(const float4*)(box_pred + ((size_t)b * NANCH + idx) * 4);
      float px = (t.x * 0.1f) * aw + cx;
      float py = (t.y * 0.1f) * ah + cy;
      float pw = __expf(t.z * 0.2f) * aw;
      float ph = __expf(t.w * 0.2f) * ah;
      float x1 = px - 0.5f * pw, y1 = py - 0.5f * ph;
      float x2 = px + 0.5f * pw, y2 = py + 0.5f * ph;
      s_boxes[tid * 4 + 0] = x1; s_boxes[tid * 4 + 1] = y1;
      s_boxes[tid * 4 + 2] = x2; s_boxes[tid * 4 + 3] = y2;
      s_area[tid] = (x2 - x1) * (y2 - y1);
      s_keep[tid] = (s_selv[tid] > 0.0f) ? 1 : 0;
    } else {
      s_boxes[tid * 4 + 0] = 0.f; s_boxes[tid * 4 + 1] = 0.f;
      s_boxes[tid * 4 + 2] = 0.f; s_boxes[tid * 4 + 3] = 0.f;
      s_area[tid] = 0.f;
      s_keep[tid] = 0;
    }
  }
  __syncthreads();

  // ---- IoU: union outer-sum via V_WMMA_F32_16X16X4_F32, intersection via VALU ----
  for (int t = wid; t < 49; t += 8) {       // wave-uniform tile loop, EXEC all-ones
    const int tr = t / 7, tc = t - 7 * tr;
    const int i0 = tr * 16, j0 = tc * 16;
    const bool lo = (lane < 16);
    const float aA = s_area[i0 + (lane & 15)];
    const float aB = s_area[j0 + (lane & 15)];
    v2f Av, Bv;
    Av.x = lo ? aA  : 0.f;  Av.y = lo ? 1.f : 0.f;   // A 16x4: col0=areaA, col1=1
    Bv.x = lo ? 1.f : 0.f;  Bv.y = lo ? aB  : 0.f;   // B 4x16: row0=1,     row1=areaB
    v8f D;
#if defined(HAS_WMMA4)
    v8f Cz = {0.f, 0.f, 0.f, 0.f, 0.f, 0.f, 0.f, 0.f};
    D = __builtin_amdgcn_wmma_f32_16x16x4_f32(false, Av, false, Bv,
                                              (short)0, Cz, false, false);
#else
#pragma unroll
    for (int r = 0; r < 8; ++r) {
      int ii = i0 + r + (lo ? 0 : 8);
      int jj = j0 + (lane & 15);
      D[r] = s_area[ii] + s_area[jj];
    }
#endif
#pragma unroll
    for (int r = 0; r < 8; ++r) {
      const int i = i0 + r + (lo ? 0 : 8);
      const int j = j0 + (lane & 15);
      if (i < KTOP && j < KTOP) {
        float xa1 = s_boxes[i*4+0], ya1 = s_boxes[i*4+1];
        float xa2 = s_boxes[i*4+2], ya2 = s_boxes[i*4+3];
        float xb1 = s_boxes[j*4+0], yb1 = s_boxes[j*4+1];
        float xb2 = s_boxes[j*4+2], yb2 = s_boxes[j*4+3];
        float iw = fmaxf(fminf(xa2, xb2) - fmaxf(xa1, xb1), 0.f);
        float ih = fmaxf(fminf(ya2, yb2) - fmaxf(ya1, yb1), 0.f);
        float inter = iw * ih;
        float uni = D[r] - inter;                 // areaA + areaB - inter
        s_iou[i * 112 + j] = inter / (uni + 1e-8f);
      }
    }
  }
  __syncthreads();

  // ---- greedy suppression (matches reference scan order) ----
  for (int i = 0; i < KTOP; ++i) {
    __syncthreads();
    if (s_keep[i]) {
      for (int j = i + 1 + tid; j < KTOP; j += nt) {
        if (s_iou[i * 112 + j] > 0.5f) s_keep[j] = 0;
      }
    }
  }
  __syncthreads();

  // ---- emit per-class candidates ----
  if (tid < KTOP) {
    const float key = s_selv[tid];
    const float score = (s_keep[tid] && key > 0.f)
                      ? (1.0f / (1.0f + __expf(-key))) : -1.0f;
    const size_t o = ((size_t)b * NUM_CLASSES + c) * KTOP + tid;
    cand_score[o] = score;
    float4 bx;
    bx.x = s_boxes[tid*4+0]; bx.y = s_boxes[tid*4+1];
    bx.z = s_boxes[tid*4+2]; bx.w = s_boxes[tid*4+3];
    ((float4*)cand_box)[o] = bx;
  }
}

// One workgroup per batch: tournament top-100 over 2000 candidates, emit outputs.
__global__ __launch_bounds__(256) void retina_combine(
    const float* __restrict__ cand_box, const float* __restrict__ cand_score,
    float* __restrict__ out, int B)
{
  __shared__ float s2[2048];
  __shared__ float s_gv2[32];
  __shared__ int   s_gi2[32];
  __shared__ float s_selv[128];
  __shared__ int   s_seli[128];
  __shared__ int   s_cnt;

  const int b = blockIdx.x, tid = threadIdx.x, nt = 256;
  const int lane = tid & 31, wid = tid >> 5;
  const int M = NUM_CLASSES * KTOP;   // 2000; flat idx = c*100+k (ref order)

  for (int i = tid; i < 2048; i += nt)
    s2[i] = (i < M) ? cand_score[(size_t)b * M + i] : SENT;
  __syncthreads();

  // group maxima: one wave per group of 64
  for (int g = wid; g < 32; g += 8) {
    const int g0 = g * 64;
    const int i0 = g0 + lane, i1 = g0 + 32 + lane;
    float v0 = s2[i0], v1 = s2[i1];
    float nv; int ni;
    if (v0 >= v1) { nv = v0; ni = i0; } else { nv = v1; ni = i1; }
    wave_argmax2(nv, ni);
    if (lane == 0) { s_gv2[g] = nv; s_gi2[g] = ni; }
  }
  __syncthreads();

  if (wid == 0) {
    for (int k = 0; k < MAXDET; ++k) {
      float bv = s_gv2[lane]; int bi = s_gi2[lane];
      wave_argmax2(bv, bi);
      bv = __shfl(bv, 0, 32);
      bi = __shfl(bi, 0, 32);
      if (lane == 0) {
        s_selv[k] = bv; s_seli[k] = bi;
        if (bi < 2048) s2[bi] = SENT;
      }
      const int g0 = (bi < 2048) ? (bi >> 6) << 6 : 0;
      const int i0 = g0 + lane, i1 = g0 + 32 + lane;
      float v0 = s2[i0], v1 = s2[i1];
      float nv; int ni;
      if (v0 >= v1) { nv = v0; ni = i0; } else { nv = v1; ni = i1; }
      wave_argmax2(nv, ni);
      if (lane == 0 && bi < 2048) { s_gv2[g0 >> 6] = nv; s_gi2[g0 >> 6] = ni; }
    }
  }
  __syncthreads();

  if (tid == 0) s_cnt = 0;
  __syncthreads();

  if (tid < MAXDET) {
    const float fs = s_selv[tid];
    const int   fi = s_seli[tid];
    const bool valid = (fs > 0.0f);
    float4 bx; bx.x = 0.f; bx.y = 0.f; bx.z = 0.f; bx.w = 0.f;
    if (valid && fi < M) bx = ((const float4*)cand_box)[(size_t)b * M + fi];
    float* ob = out + ((size_t)b * MAXDET + tid) * 4;
    ob[0] = bx.x; ob[1] = bx.y; ob[2] = bx.z; ob[3] = bx.w;
    out[(size_t)B * MAXDET * 4 + (size_t)b * MAXDET + tid] = valid ? fs : 0.f;
    out[(size_t)B * MAXDET * 5 + (size_t)b * MAXDET + tid] =
        valid ? (float)(fi / KTOP) : 0.f;
    if (valid) atomicAdd(&s_cnt, 1);
  }
  __syncthreads();
  if (tid == 0) out[(size_t)B * MAXDET * 6 + b] = (float)s_cnt;
}

extern "C" void kernel_launch(void* const* d_in, const int* in_sizes, int n_in,
                              void* d_out, int out_size, void* d_ws, size_t ws_size,
                              hipStream_t stream) {
  (void)n_in; (void)out_size; (void)ws_size;
  // d_in[0]=images (unused, shape only), d_in[1]=box_pred, d_in[2]=cls_pred
  const float* box_pred = (const float*)d_in[1];
  const float* cls_pred = (const float*)d_in[2];
  float* out = (float*)d_out;
  const int B = in_sizes[1] / (NANCH * 4);

  float* cand_box   = (float*)d_ws;  // [B,20,100,4]
  float* cand_score = (float*)((char*)d_ws +
                      (size_t)B * NUM_CLASSES * KTOP * 4 * sizeof(float));

  dim3 gA(NUM_CLASSES, B);
  retina_perclass_nms<<<gA, 256, 0, stream>>>(box_pred, cls_pred,
                                              cand_box, cand_score);
  retina_combine<<<B, 256, 0, stream>>>(cand_box, cand_score, out, B);
}